// TransducerLoss_14061722927180
// MI455X (gfx1250) — compile-verified
//
#include <hip/hip_runtime.h>

typedef float v2f __attribute__((ext_vector_type(2)));
typedef float v4f __attribute__((ext_vector_type(4)));
typedef float v8f __attribute__((ext_vector_type(8)));

#define BB 16
#define TT 192
#define UU 64
#define HH 512

// ---------------------------------------------------------------------------
// Kernel 1: per (b,t), compute denom = logsumexp_H via WMMA row-sums of exp,
// then lp_blank[b,t,u], lp_label -> cs[b,t,0..63] (cumsum over u).
// Grid: (T, B), block 128 threads = 4 waves; wave w owns rows u = w*16 .. +15.
// ---------------------------------------------------------------------------
__global__ __launch_bounds__(128)
void k_lse_wmma(const float* __restrict__ x, const int* __restrict__ label,
                const int* __restrict__ blank_ptr,
                float* __restrict__ lpb, float* __restrict__ cs)
{
    // pitch 36 floats: WMMA A-layout ds_load_b64 pattern hits 64 distinct banks
    __shared__ float tile[4][16][36];
    __shared__ float lpl[UU];

    const int t    = blockIdx.x;
    const int b    = blockIdx.y;
    const int tid  = threadIdx.x;
    const int wave = tid >> 5;
    const int lane = tid & 31;
    const int u0   = wave * 16;
    const int blank = *blank_ptr;

    const size_t tileBase = ((size_t)(b * TT + t) * UU + u0) * HH;

    const v2f ones = {1.0f, 1.0f};
    v8f acc = {0.f, 0.f, 0.f, 0.f, 0.f, 0.f, 0.f, 0.f};

    // global-load roles: 8 lanes per row -> 128B contiguous per row, 4 rows/instr
    const int lrow = lane >> 3;        // 0..3
    const int lcol = (lane & 7) * 4;   // 0,4,...,28
    float (*my)[36] = tile[wave];

    // A-layout read roles (V_WMMA_F32_16X16X4_F32):
    // lanes 0-15: M=lane, K={0,1}; lanes 16-31: M=lane-16, K={2,3}
    const int arow = lane & 15;
    const int aoff = (lane >> 4) * 2;

    // prologue load of chunk c=0
    v4f ld[4];
    #pragma unroll
    for (int i = 0; i < 4; ++i) {
        const int r = i * 4 + lrow;
        ld[i] = *(const v4f*)(x + tileBase + (size_t)r * HH + lcol);
    }

    for (int c = 0; c < HH; c += 32) {
        // stage current chunk to LDS (two b64 stores keep 8B alignment w/ pitch 36)
        #pragma unroll
        for (int i = 0; i < 4; ++i) {
            const int r = i * 4 + lrow;
            *(v2f*)&my[r][lcol]     = (v2f){ld[i].x, ld[i].y};
            *(v2f*)&my[r][lcol + 2] = (v2f){ld[i].z, ld[i].w};
        }
        // issue next chunk's global loads (consumed next iteration)
        if (c + 32 < HH) {
            #pragma unroll
            for (int i = 0; i < 4; ++i) {
                const int r = i * 4 + lrow;
                ld[i] = *(const v4f*)(x + tileBase + (size_t)r * HH + (c + 32) + lcol);
            }
        }
        __builtin_amdgcn_wave_barrier();   // keep ds_store before ds_load
        // 8 x WMMA 16x16x4: row-sums of exp() accumulate into acc
        #pragma unroll
        for (int k = 0; k < 8; ++k) {
            v2f a = *(const v2f*)&my[arow][k * 4 + aoff];
            a.x = __expf(a.x);
            a.y = __expf(a.y);
            acc = __builtin_amdgcn_wmma_f32_16x16x4_f32(
                false, a, false, ones, (short)0, acc, false, false);
        }
        __builtin_amdgcn_wave_barrier();   // keep next iter's stores after loads
    }

    // C/D layout: lanes 0-15 hold M=r in vgpr r; lanes 16-31 hold M=r+8.
    // 16 lanes (0..7 and 16..23) each finalize one row.
    if ((lane & 15) < 8) {
        const int r = lane & 7;
        const float s =
            (r == 0) ? acc[0] : (r == 1) ? acc[1] : (r == 2) ? acc[2] :
            (r == 3) ? acc[3] : (r == 4) ? acc[4] : (r == 5) ? acc[5] :
            (r == 6) ? acc[6] : acc[7];
        const int row = r + (lane >> 4) * 8;
        const int u   = u0 + row;
        const float denom = __logf(s);
        const size_t rbase = ((size_t)(b * TT + t) * UU + u) * HH;
        lpb[(size_t)(b * TT + t) * UU + u] = x[rbase + blank] - denom;
        if (u < UU - 1) {
            const int lab = label[b * (UU - 1) + u];
            lpl[u] = x[rbase + lab] - denom;
        }
    }
    __syncthreads();

    if (tid == 0) {
        float* csrow = cs + (size_t)(b * TT + t) * UU;
        float run = 0.f;
        csrow[0] = 0.f;
        for (int u = 0; u < UU - 1; ++u) {
            run += lpl[u];
            csrow[u + 1] = run;
        }
    }
}

// ---------------------------------------------------------------------------
// Kernel 2: alpha recursion. One wave per batch; lane l holds u=2l, 2l+1.
// cumlogsumexp over U=64 = lane-pair combine + 5-step wave32 inclusive scan.
// ---------------------------------------------------------------------------
__device__ __forceinline__ float lse2(float a, float b) {
    const float m = fmaxf(a, b);
    const float d = fminf(a, b) - m;          // <= 0; exp(-inf)=0 handles -inf
    return m + __logf(1.0f + __expf(d));
}

__global__ __launch_bounds__(32)
void k_alpha(const float* __restrict__ lpb, const float* __restrict__ cs,
             const int* __restrict__ f_len, const int* __restrict__ y_len,
             float* __restrict__ out)
{
    const int b    = blockIdx.x;
    const int lane = threadIdx.x;
    const int tl   = f_len[b] - 1;
    const int ul   = y_len[b];
    const float NEG_INF = -__builtin_inff();

    const float* csb  = cs  + (size_t)b * TT * UU;
    const float* lpbb = lpb + (size_t)b * TT * UU;

    v2f a01 = *(const v2f*)(csb + 2 * lane);      // alpha0 = cs[:,0,:]
    float a0 = a01.x, a1 = a01.y;

    if (tl == 0) {
        if (2 * lane == ul)     out[b] = -(a0 + lpbb[(size_t)tl * UU + ul]);
        if (2 * lane + 1 == ul) out[b] = -(a1 + lpbb[(size_t)tl * UU + ul]);
    }

    for (int t = 1; t < TT; ++t) {
        const v2f lb = *(const v2f*)(lpbb + (size_t)(t - 1) * UU + 2 * lane);
        const v2f ct = *(const v2f*)(csb  + (size_t)t       * UU + 2 * lane);
        const float e0 = a0 + lb.x - ct.x;
        const float e1 = a1 + lb.y - ct.y;

        float s = lse2(e0, e1);                   // pair-local combine
        #pragma unroll
        for (int off = 1; off < 32; off <<= 1) {  // inclusive wave scan (lse)
            const float o = __shfl_up(s, off, 32);
            if (lane >= off) s = lse2(s, o);
        }
        float p = __shfl_up(s, 1, 32);            // exclusive prefix
        if (lane == 0) p = NEG_INF;

        a0 = ct.x + lse2(p, e0);                  // cumlse at u=2l
        a1 = ct.y + s;                            // cumlse at u=2l+1

        if (t == tl) {
            if (2 * lane == ul)
                out[b] = -(a0 + lpbb[(size_t)tl * UU + ul]);
            else if (2 * lane + 1 == ul)
                out[b] = -(a1 + lpbb[(size_t)tl * UU + ul]);
        }
    }
}

// ---------------------------------------------------------------------------
extern "C" void kernel_launch(void* const* d_in, const int* in_sizes, int n_in,
                              void* d_out, int out_size, void* d_ws, size_t ws_size,
                              hipStream_t stream)
{
    const float* x     = (const float*)d_in[0];
    const int*   label = (const int*)d_in[1];
    const int*   f_len = (const int*)d_in[2];
    const int*   y_len = (const int*)d_in[3];
    const int*   blank = (const int*)d_in[4];

    float* lpb = (float*)d_ws;                       // B*T*U floats
    float* cs  = lpb + (size_t)BB * TT * UU;         // B*T*U floats

    k_lse_wmma<<<dim3(TT, BB), 128, 0, stream>>>(x, label, blank, lpb, cs);
    k_alpha<<<BB, 32, 0, stream>>>(lpb, cs, f_len, y_len, (float*)d_out);
}